// MaskingModule_89094801588989
// MI455X (gfx1250) — compile-verified
//
#include <hip/hip_runtime.h>
#include <stdint.h>

// ---------------------------------------------------------------------------
// MaskingModule for MI455X (gfx1250, wave32).
//   x: (32, 4096, 768) fp32
//   outputs (concatenated float): x_masked (32*1024*768), mask (32*4096),
//                                 ids_restore (32*4096, stored as float)
// Pipeline:
//   k_entropy: async global->LDS staged (GLOBAL_LOAD_ASYNC_TO_LDS_B128,
//              ASYNCcnt), wave-per-token 10-bin histogram entropy via
//              wave32 ballots + ds_swizzle min/max reduction.
//   k_rank:    per-row stable rank (== argsort(-ent) inverse permutation).
//   k_gather:  float4 gather of kept tokens.
// ---------------------------------------------------------------------------

constexpr int kN    = 32;
constexpr int kL    = 4096;
constexpr int kD    = 768;     // floats per token (3072 bytes)
constexpr int kKeep = 1024;    // L * (1 - 0.75)
constexpr int kBins = 10;

// ---- CDNA5 async global->LDS availability probes --------------------------
#if defined(__gfx1250__) && __has_builtin(__builtin_amdgcn_global_load_async_to_lds_b128)
#define HAVE_ASYNC_LDS 1
#else
#define HAVE_ASYNC_LDS 0
#endif

#if defined(__gfx1250__) && __has_builtin(__builtin_amdgcn_s_wait_asynccnt)
#define HAVE_WAIT_ASYNC 1
#else
#define HAVE_WAIT_ASYNC 0
#endif

// Vector-of-4-int type matching the builtin's parameter type
// ('__attribute__((__vector_size__(4 * sizeof(int)))) int *').
typedef int v4i_t __attribute__((vector_size(16)));
typedef __attribute__((address_space(1))) v4i_t* glob_v4i_ptr;
typedef __attribute__((address_space(3))) v4i_t* lds_v4i_ptr;

// ---- wave32 helpers --------------------------------------------------------
// ds_swizzle group-of-32 xor: offset = {1'b0, xor[4:0], or[4:0]=0, and[4:0]=0x1f}
template <int XorMask>
__device__ __forceinline__ float swz_xor(float v) {
  return __int_as_float(
      __builtin_amdgcn_ds_swizzle(__float_as_int(v), (XorMask << 10) | 0x1f));
}

template <int NRemain>
__device__ __forceinline__ void wait_async_le() {
#if HAVE_ASYNC_LDS
#if HAVE_WAIT_ASYNC
  __builtin_amdgcn_s_wait_asynccnt(NRemain);
  // Compiler-level barrier: keep the LDS consumer reads below the wait.
  asm volatile("" ::: "memory");
#else
  asm volatile("s_wait_asynccnt %0" ::"n"(NRemain) : "memory");
#endif
#endif
}

// Stage one 768-float token tile into LDS. 6 x (32 lanes x 16B) = 3072B.
// Each lane writes the same LDS bytes it later reads (lane*16 + c*512).
__device__ __forceinline__ void load_token(const float* __restrict__ g,
                                           float* l, int lane) {
#if HAVE_ASYNC_LDS
  const float* gp = g + lane * 4;
  float*       lp = l + lane * 4;
#pragma unroll
  for (int c = 0; c < 6; ++c) {
    __builtin_amdgcn_global_load_async_to_lds_b128(
        (glob_v4i_ptr)(gp + c * 128),
        (lds_v4i_ptr)(lp + c * 128),
        /*offset=*/0, /*cpol=*/0);
  }
#else
  const float4* g4 = (const float4*)g;
  float4*       l4 = (float4*)l;
#pragma unroll
  for (int c = 0; c < 6; ++c) l4[lane + 32 * c] = g4[lane + 32 * c];
#endif
}

// Whole wave computes the entropy of one token held in LDS (uniform result).
__device__ __forceinline__ float token_entropy_wave(const float* lbuf, int lane) {
  float v[24];
  const float4* b4 = (const float4*)lbuf;
#pragma unroll
  for (int c = 0; c < 6; ++c) {
    float4 q = b4[lane + 32 * c];
    v[4 * c + 0] = q.x; v[4 * c + 1] = q.y;
    v[4 * c + 2] = q.z; v[4 * c + 3] = q.w;
  }
  float mn = v[0], mx = v[0];
#pragma unroll
  for (int j = 1; j < 24; ++j) { mn = fminf(mn, v[j]); mx = fmaxf(mx, v[j]); }
  // wave32 xor-swizzle min/max reduction (all 32 lanes active).
  mn = fminf(mn, swz_xor<1>(mn));   mx = fmaxf(mx, swz_xor<1>(mx));
  mn = fminf(mn, swz_xor<2>(mn));   mx = fmaxf(mx, swz_xor<2>(mx));
  mn = fminf(mn, swz_xor<4>(mn));   mx = fmaxf(mx, swz_xor<4>(mx));
  mn = fminf(mn, swz_xor<8>(mn));   mx = fmaxf(mx, swz_xor<8>(mx));
  mn = fminf(mn, swz_xor<16>(mn));  mx = fmaxf(mx, swz_xor<16>(mx));

  const float inv = 1.0f / ((mx - mn) + 1e-19f);  // EPS_NORM
  int cnt[kBins] = {};
#pragma unroll
  for (int j = 0; j < 24; ++j) {
    float z = (v[j] - mn) * inv;
    int   q = (int)floorf(z * (float)(kBins - 1));
    q = q > (kBins - 1) ? (kBins - 1) : q;
    // wave-ballot histogram: 1 cmp + popc per (value, bin); counts uniform.
#pragma unroll
    for (int b = 0; b < kBins; ++b)
      cnt[b] += __popc(__builtin_amdgcn_ballot_w32(q == b));
  }
  float e = 0.0f;
#pragma unroll
  for (int b = 0; b < kBins; ++b) {
    float p = (float)cnt[b] / (float)kD;
    e -= p * logf(p + 1e-9f);  // EPS_LOG
  }
  return e;
}

// 8 waves/block, 8 tokens/wave, double-buffered async prefetch.
__global__ __launch_bounds__(256) void k_entropy(const float* __restrict__ x,
                                                 float* __restrict__ ent) {
  __shared__ __align__(16) float buf[8][2][kD];  // 48 KB
  const int  w    = threadIdx.x >> 5;
  const int  lane = threadIdx.x & 31;
  const long long t0 = ((long long)blockIdx.x * 8 + w) * 8;

  load_token(x + t0 * kD, &buf[w][0][0], lane);
  // Keep rolled: body is large (240 ballots); unrolling caused SGPR spills.
#pragma unroll 1
  for (int k = 0; k < 8; ++k) {
    if (k + 1 < 8) {
      load_token(x + (t0 + k + 1) * kD, &buf[w][(k + 1) & 1][0], lane);
      wait_async_le<6>();   // in-order completion: token k's 6 chunks are done
    } else {
      wait_async_le<0>();
    }
    const float e = token_entropy_wave(&buf[w][k & 1][0], lane);
    if (lane == 0) ent[t0 + k] = e;
  }
}

// Stable descending rank == inverse of jnp.argsort(-ent) (stable ties by idx).
// 4 blocks of 1024 threads per row; row entropies cached in LDS.
__global__ __launch_bounds__(1024) void k_rank(const float* __restrict__ ent,
                                               float* __restrict__ mask_out,
                                               float* __restrict__ ids_out,
                                               int* __restrict__ keep_src) {
  __shared__ __align__(16) float se[kL];  // 16 KB
  const int n   = blockIdx.x >> 2;
  const int seg = blockIdx.x & 3;
  const float* rowe = ent + (size_t)n * kL;
  for (int i = threadIdx.x; i < kL; i += 1024) se[i] = rowe[i];
  __syncthreads();

  const int   i = seg * 1024 + threadIdx.x;
  const float e = se[i];
  int rank = 0;
  const float4* se4 = (const float4*)se;
  for (int j4 = 0; j4 < kL / 4; ++j4) {
    const float4 vv = se4[j4];
    const int    jb = j4 * 4;
    rank += (int)(vv.x > e) + (int)((vv.x == e) && (jb + 0 < i));
    rank += (int)(vv.y > e) + (int)((vv.y == e) && (jb + 1 < i));
    rank += (int)(vv.z > e) + (int)((vv.z == e) && (jb + 2 < i));
    rank += (int)(vv.w > e) + (int)((vv.w == e) && (jb + 3 < i));
  }
  const int g = n * kL + i;
  ids_out[g]  = (float)rank;                       // ids_restore
  mask_out[g] = (rank < kKeep) ? 0.0f : 1.0f;      // mask
  if (rank < kKeep) keep_src[n * kKeep + rank] = i;
}

// x_masked[n, r, :] = x[n, keep_src[n*kKeep + r], :]; one block per output row.
__global__ __launch_bounds__(192) void k_gather(const float* __restrict__ x,
                                                const int* __restrict__ keep_src,
                                                float* __restrict__ xm) {
  const int row = blockIdx.x;         // n*kKeep + r
  const int n   = row >> 10;
  const int src = keep_src[row];      // uniform -> scalar load
  const float4* s4 = (const float4*)(x + ((size_t)n * kL + src) * kD);
  float4*       d4 = (float4*)(xm + (size_t)row * kD);
  d4[threadIdx.x] = s4[threadIdx.x];
}

// ---------------------------------------------------------------------------
extern "C" void kernel_launch(void* const* d_in, const int* in_sizes, int n_in,
                              void* d_out, int out_size, void* d_ws, size_t ws_size,
                              hipStream_t stream) {
  (void)in_sizes; (void)n_in; (void)out_size; (void)ws_size;
  const float* x = (const float*)d_in[0];

  float* xm   = (float*)d_out;                          // 32*1024*768
  float* mask = xm + (size_t)kN * kKeep * kD;           // 32*4096
  float* ids  = mask + (size_t)kN * kL;                 // 32*4096

  float* ent  = (float*)d_ws;                                          // 512 KB
  int*   keep = (int*)((char*)d_ws + sizeof(float) * (size_t)kN * kL); // 128 KB

  k_entropy<<<(kN * kL) / 64, 256, 0, stream>>>(x, ent);
  k_rank<<<kN * (kL / 1024), 1024, 0, stream>>>(ent, mask, ids, keep);
  k_gather<<<kN * kKeep, 192, 0, stream>>>(x, keep, xm);
}